// QuanvolutionFilter_65481071404025
// MI455X (gfx1250) — compile-verified
//
#include <hip/hip_runtime.h>

// ---------------------------------------------------------------------------
// Quanvolution circuit on MI455X (gfx1250).
//
// Roofline: 25.7MB read + 25.7MB write ~= 51MB -> ~2.2us at 23.3 TB/s.
// The fixed 8-gate circuit collapses to one 16x16 complex unitary U
// (depends only on rand_params), so per patch: psi = U * psi0 with psi0 REAL.
// We batch 16 patches per wave and evaluate Re/Im parts with
// V_WMMA_F32_16X16X4_F32 chained over 4 K-chunks (8 WMMAs / 16 patches),
// then reduce |psi|^2 with Z-sign masks + one cross-half shuffle.
// ---------------------------------------------------------------------------

typedef __attribute__((ext_vector_type(2))) float v2f;
typedef __attribute__((ext_vector_type(8))) float v8f;

// --- Setup kernel: build U (16x16 complex) from the 5 rand_params. ---------
// Thread j computes column j = circuit(e_j).  Wire w <-> bit (3-w) of index.
__global__ void build_unitary_kernel(const float* __restrict__ p,
                                     float* __restrict__ Ur,
                                     float* __restrict__ Ui) {
  const int j = threadIdx.x;
  if (j >= 16) return;

  float vr[16], vi[16];
#pragma unroll
  for (int i = 0; i < 16; ++i) { vr[i] = (i == j) ? 1.0f : 0.0f; vi[i] = 0.0f; }

  auto apply1q = [&](int mask,
                     float a00r, float a00i, float a01r, float a01i,
                     float a10r, float a10i, float a11r, float a11i) {
#pragma unroll
    for (int i = 0; i < 16; ++i) {
      if ((i & mask) == 0) {
        const int i1 = i | mask;
        const float xr = vr[i],  xi = vi[i];
        const float yr = vr[i1], yi = vi[i1];
        vr[i]  = a00r*xr - a00i*xi + a01r*yr - a01i*yi;
        vi[i]  = a00r*xi + a00i*xr + a01r*yi + a01i*yr;
        vr[i1] = a10r*xr - a10i*xi + a11r*yr - a11i*yi;
        vi[i1] = a10r*xi + a10i*xr + a11r*yi + a11i*yr;
      }
    }
  };
  auto cnot = [&](int cmask, int tmask) {   // new[i] = old[i ^ tmask] if control set
#pragma unroll
    for (int i = 0; i < 16; ++i) {
      if ((i & cmask) && !(i & tmask)) {
        const int i1 = i | tmask;
        const float tr = vr[i], ti = vi[i];
        vr[i] = vr[i1]; vi[i] = vi[i1];
        vr[i1] = tr;    vi[i1] = ti;
      }
    }
  };

  float c, s;
  c = cosf(p[0]*0.5f); s = sinf(p[0]*0.5f);                 // RX on wire0 (mask 8)
  apply1q(8, c,0.f,  0.f,-s,  0.f,-s,  c,0.f);
  c = cosf(p[1]*0.5f); s = sinf(p[1]*0.5f);                 // RY on wire1 (mask 4)
  apply1q(4, c,0.f,  -s,0.f,  s,0.f,   c,0.f);
  c = cosf(p[2]*0.5f); s = sinf(p[2]*0.5f);                 // RZ on wire2 (mask 2)
  apply1q(2, c,-s,   0.f,0.f, 0.f,0.f, c,s);
  cnot(8, 4);                                               // CNOT(0,1)
  c = cosf(p[3]*0.5f); s = sinf(p[3]*0.5f);                 // RY on wire3 (mask 1)
  apply1q(1, c,0.f,  -s,0.f,  s,0.f,   c,0.f);
  cnot(2, 1);                                               // CNOT(2,3)
  c = cosf(p[4]*0.5f); s = sinf(p[4]*0.5f);                 // RX on wire2 (mask 2)
  apply1q(2, c,0.f,  0.f,-s,  0.f,-s,  c,0.f);
  cnot(4, 2);                                               // CNOT(1,2)

#pragma unroll
  for (int i = 0; i < 16; ++i) {
    Ur[i*16 + j] = vr[i];   // row-major: U[out][in]
    Ui[i*16 + j] = vi[i];
  }
}

// --- Main kernel: one wave == 16 patches via WMMA f32 16x16x4. -------------
__global__ __launch_bounds__(256)
void quanv_wmma_kernel(const float* __restrict__ x,
                       const float* __restrict__ Ur,
                       const float* __restrict__ Ui,
                       float* __restrict__ out, int n_patches) {
  const int lane  = threadIdx.x & 31;
  const int wave  = threadIdx.x >> 5;
  const int gwave = blockIdx.x * 8 + wave;
  const int base  = gwave * 16;
  if (base >= n_patches) return;          // wave-uniform: EXEC stays all-ones

  const int half  = lane >> 4;            // 0: K={4c,4c+1}  1: K={4c+2,4c+3}
  const int col   = lane & 15;            // A row / B column / D column
  const int patch = base + col;

  // A operands: per-lane float2 slices of U (16x4 f32 A layout).
  const int koff = half * 2;
  v2f ar[4], ai[4];
#pragma unroll
  for (int c = 0; c < 4; ++c) {
    const int idx = col * 16 + 4*c + koff;        // 8-byte aligned
    ar[c] = *(const v2f*)(Ur + idx);
    ai[c] = *(const v2f*)(Ui + idx);
  }

  // Patch pixel angles: patch n -> b = n/196, pi = n%196, r=pi/14, cc=pi%14.
  const int b  = patch / 196;
  const int pi = patch - b * 196;
  const int r  = pi / 14;
  const int cc = pi - r * 14;
  const float* px = x + b * 784 + r * 56 + cc * 2;
  float s0, c0, s1, c1, s2, c2, s3, c3;
  __sincosf(px[0]  * 0.5f, &s0, &c0);
  __sincosf(px[1]  * 0.5f, &s1, &c1);
  __sincosf(px[28] * 0.5f, &s2, &c2);
  __sincosf(px[29] * 0.5f, &s3, &c3);

  // B operands: psi0[k] = q0[k>>3] q1[(k>>2)&1] q2[(k>>1)&1] q3[k&1].
  // chunk c covers k = 4c + 2*half + {0,1}  ->  common * {cos th3, sin th3}.
  const float f2  = half ? s2 : c2;
  const float g00 = c0 * c1 * f2;
  const float g01 = c0 * s1 * f2;
  const float g10 = s0 * c1 * f2;
  const float g11 = s0 * s1 * f2;
  v2f bm[4];
  bm[0].x = g00 * c3; bm[0].y = g00 * s3;
  bm[1].x = g01 * c3; bm[1].y = g01 * s3;
  bm[2].x = g10 * c3; bm[2].y = g10 * s3;
  bm[3].x = g11 * c3; bm[3].y = g11 * s3;

  // psi = U * psi0 : Re and Im accumulated over 4 K-chunks (interleaved for
  // pipelining; compiler inserts the WMMA->WMMA hazard delays).
  v8f dr = {};
  v8f di = {};
#pragma unroll
  for (int c = 0; c < 4; ++c) {
    dr = __builtin_amdgcn_wmma_f32_16x16x4_f32(false, ar[c], false, bm[c],
                                               (short)0, dr, false, false);
    di = __builtin_amdgcn_wmma_f32_16x16x4_f32(false, ai[c], false, bm[c],
                                               (short)0, di, false, false);
  }

  // <Z_w> = sum_m sign(bit_w(m)) * |psi_m|^2 ; this lane owns m = v + 8*half.
  float m0 = 0.f, m1 = 0.f, m2 = 0.f, m3 = 0.f;
#pragma unroll
  for (int v = 0; v < 8; ++v) {
    const float pv = dr[v]*dr[v] + di[v]*di[v];
    const int m = v + half*8;
    m0 += (m & 8) ? -pv : pv;   // wire0 = bit3
    m1 += (m & 4) ? -pv : pv;   // wire1 = bit2
    m2 += (m & 2) ? -pv : pv;   // wire2 = bit1
    m3 += (m & 1) ? -pv : pv;   // wire3 = bit0
  }
  // merge the two half-lane partial sums (lane l <-> l^16 hold same patch)
  m0 += __shfl_xor(m0, 16, 32);
  m1 += __shfl_xor(m1, 16, 32);
  m2 += __shfl_xor(m2, 16, 32);
  m3 += __shfl_xor(m3, 16, 32);

  if (half == 0) {
    float4 o; o.x = m0; o.y = m1; o.z = m2; o.w = m3;
    *(float4*)(out + (size_t)patch * 4) = o;   // 16B-aligned, coalesced
  }
}

extern "C" void kernel_launch(void* const* d_in, const int* in_sizes, int n_in,
                              void* d_out, int out_size, void* d_ws, size_t ws_size,
                              hipStream_t stream) {
  const float* x  = (const float*)d_in[0];   // [8192, 784] pixels
  const float* rp = (const float*)d_in[1];   // [5] rand_params

  float* Ur = (float*)d_ws;                  // 16x16 Re(U)
  float* Ui = Ur + 256;                      // 16x16 Im(U)

  build_unitary_kernel<<<1, 16, 0, stream>>>(rp, Ur, Ui);

  const int n_patches = out_size / 4;                // 1,605,632
  const int blocks = (n_patches + 127) / 128;        // 128 patches / 256-thr block
  quanv_wmma_kernel<<<blocks, 256, 0, stream>>>(x, Ur, Ui, (float*)d_out, n_patches);
}